// SDF2_16535624089832
// MI455X (gfx1250) — compile-verified
//
#include <hip/hip_runtime.h>
#include <math.h>

// ---------------------------------------------------------------------------
// SDF on a G^3 grid vs F triangles, CDNA5 (gfx1250, wave32).
//
// Per wave: one 16-point tile. Loop over triangles in 16-wide tiles.
// A (16x4 f32) rows = (px, py, pz, 1):
//   lanes 0-15  -> (K0,K1) = (px, py)     of point M = lane
//   lanes 16-31 -> (K2,K3) = (pz, 1.0f)   of point M = lane-16
// Eight V_WMMA_F32_16X16X4_F32 per 16-triangle tile compute, per pair:
//   d1..d6 (Ericson dot products, constants folded into B's K3 row) and the
//   two ray-cast barycentric numerators (affine in px,py).
// Remaining region-selection math is per-pair VALU in the WMMA D layout:
//   lane l, acc slot m  ->  point (m + 8*(l>=16)), triangle (l%16).
// All divisions use v_rcp_f32 (1 ulp) — quotients only position the closest
// point continuously; region selection uses raw d-values.
// Reduction over the 16 triangle columns is shfl_xor {1,2,4,8}, which stays
// inside each 16-lane half, matching the D layout.
// ---------------------------------------------------------------------------

typedef __attribute__((ext_vector_type(2))) float v2f;
typedef __attribute__((ext_vector_type(8))) float v8f;

#define EPSF   1e-12f
#define CHUNK  256          // triangles staged in LDS per pass (== blockDim)
#define TPB    256          // 8 waves -> 128 points per block

static __device__ __forceinline__ float frcp(float x) {
    return __builtin_amdgcn_rcpf(x);
}

__global__ __launch_bounds__(TPB) void sdf_wmma_kernel(
    const int*   __restrict__ faces,   // (F,3) int32
    const float* __restrict__ verts,   // (V,3) f32
    float*       __restrict__ out,     // (G,G,G) f32
    int F, int V, int G, int P)
{
    // SoA triangle records in LDS: conflict-free 16-lane consecutive reads.
    __shared__ float s_abx[CHUNK], s_aby[CHUNK], s_abz[CHUNK];
    __shared__ float s_acx[CHUNK], s_acy[CHUNK], s_acz[CHUNK];
    __shared__ float s_n1[CHUNK], s_n2[CHUNK], s_n3[CHUNK];   // -ab.a, -ac.a, -ab.b
    __shared__ float s_n4[CHUNK], s_n5[CHUNK], s_n6[CHUNK];   // -ac.b, -ab.c, -ac.c
    __shared__ float s_ax[CHUNK], s_ay[CHUNK], s_az[CHUNK];
    __shared__ float s_bx[CHUNK], s_by[CHUNK], s_bz[CHUNK];
    __shared__ float s_cx[CHUNK], s_cy[CHUNK], s_cz[CHUNK];
    __shared__ float s_e1[CHUNK], s_e2[CHUNK], s_cu[CHUNK];
    __shared__ float s_f1[CHUNK], s_f2[CHUNK], s_cv[CHUNK];
    __shared__ float s_iv[CHUNK], s_nd[CHUNK];

    const int  tid  = threadIdx.x;
    const int  lane = tid & 31;
    const int  wave = tid >> 5;
    const bool hi   = lane >= 16;
    const int  n16  = lane & 15;

    const int tileBase = blockIdx.x * 128 + wave * 16;
    const float invG = 1.0f / (float)G;

    // --- A matrix: rows (px, py, pz, 1)
    float pax = 0.f, pay = 0.f, paz = 0.f;
    {
        int pid = tileBase + n16;
        if (pid < P) {
            int ix = pid / (G * G), iy = (pid / G) % G, iz = pid % G;
            pax = ((float)ix + 0.5f) * invG;
            pay = ((float)iy + 0.5f) * invG;
            paz = ((float)iz + 0.5f) * invG;
        }
    }
    v2f A;
    A.x = hi ? paz : pax;
    A.y = hi ? 1.0f : pay;

    // --- per-accumulator-slot point coords (slot m -> point tileBase+m+8*hi)
    float pxs[8], pys[8], pzs[8], md2[8];
    int   par[8];
#pragma unroll
    for (int m = 0; m < 8; ++m) {
        int pid = tileBase + m + (hi ? 8 : 0);
        int q   = (pid < P) ? pid : 0;
        int ix = q / (G * G), iy = (q / G) % G, iz = q % G;
        pxs[m] = ((float)ix + 0.5f) * invG;
        pys[m] = ((float)iy + 0.5f) * invG;
        pzs[m] = ((float)iz + 0.5f) * invG;
        md2[m] = __builtin_inff();
        par[m] = 0;
    }

    for (int cb = 0; cb < F; cb += CHUNK) {
        __syncthreads();
        // ---------------- stage one chunk of triangle records -------------
        {
            int t = cb + tid;
            // gfx1250 prefetch of next chunk's face indices (speculative)
            if (cb + CHUNK < F)
                __builtin_prefetch(&faces[3 * (t + CHUNK)], 0, 3);
            float ax, ay, az, bx, by, bz, cx, cy, cz;
            if (t < F) {
                int f0 = faces[3 * t + 0], f1 = faces[3 * t + 1], f2 = faces[3 * t + 2];
                ax = verts[3 * f0 + 0]; ay = verts[3 * f0 + 1]; az = verts[3 * f0 + 2];
                bx = verts[3 * f1 + 0]; by = verts[3 * f1 + 1]; bz = verts[3 * f1 + 2];
                cx = verts[3 * f2 + 0]; cy = verts[3 * f2 + 1]; cz = verts[3 * f2 + 2];
            } else {
                // padding triangle: huge distance, never crosses the ray
                ax = ay = az = bx = by = bz = cx = cy = cz = 1e18f;
            }
            float abx = bx - ax, aby = by - ay, abz = bz - az;
            float acx = cx - ax, acy = cy - ay, acz = cz - az;
            s_abx[tid] = abx; s_aby[tid] = aby; s_abz[tid] = abz;
            s_acx[tid] = acx; s_acy[tid] = acy; s_acz[tid] = acz;
            s_n1[tid] = -(abx * ax + aby * ay + abz * az);   // -ab.a
            s_n2[tid] = -(acx * ax + acy * ay + acz * az);   // -ac.a
            s_n3[tid] = -(abx * bx + aby * by + abz * bz);   // -ab.b
            s_n4[tid] = -(acx * bx + acy * by + acz * bz);   // -ac.b
            s_n5[tid] = -(abx * cx + aby * cy + abz * cz);   // -ab.c
            s_n6[tid] = -(acx * cx + acy * cy + acz * cz);   // -ac.c
            s_ax[tid] = ax; s_ay[tid] = ay; s_az[tid] = az;
            s_bx[tid] = bx; s_by[tid] = by; s_bz[tid] = bz;
            s_cx[tid] = cx; s_cy[tid] = cy; s_cz[tid] = cz;
            // ray crossing: u = (e1*px + e2*py + Cu)/safe, v = (f1*px + f2*py + Cv)/safe
            float e1 = by - cy, e2 = cx - bx;
            float f1 = cy - ay, f2 = ax - cx;
            float dn = e1 * (ax - cx) + e2 * (ay - cy);
            float nd = (fabsf(dn) >= EPSF) ? 1.0f : 0.0f;
            float sf = (nd > 0.5f) ? dn : EPSF;
            s_e1[tid] = e1; s_e2[tid] = e2; s_cu[tid] = -e1 * cx - e2 * cy;
            s_f1[tid] = f1; s_f2[tid] = f2; s_cv[tid] = -f1 * cx - f2 * cy;
            s_iv[tid] = (t < F) ? (1.0f / sf) : 0.0f;
            s_nd[tid] = (t < F) ? nd : 0.0f;
        }
        __syncthreads();

        // ---------------- 16 triangle tiles per chunk ----------------------
        for (int it = 0; it < CHUNK / 16; ++it) {
            const int n = it * 16 + n16;   // lanes l and l+16 read same record (broadcast)
            const float abx = s_abx[n], aby = s_aby[n], abz = s_abz[n];
            const float acx = s_acx[n], acy = s_acy[n], acz = s_acz[n];
            const float nc1 = s_n1[n], nc2 = s_n2[n], nc3 = s_n3[n];
            const float nc4 = s_n4[n], nc5 = s_n5[n], nc6 = s_n6[n];
            const float ax = s_ax[n], ay = s_ay[n], az = s_az[n];
            const float bx = s_bx[n], by = s_by[n], bz = s_bz[n];
            const float cx = s_cx[n], cy = s_cy[n], cz = s_cz[n];
            const float re1 = s_e1[n], re2 = s_e2[n], rcu = s_cu[n];
            const float rf1 = s_f1[n], rf2 = s_f2[n], rcv = s_cv[n];
            const float riv = s_iv[n], rnd = s_nd[n];
            const float bcx = cx - bx, bcy = cy - by, bcz = cz - bz;

            // B columns (4x16 f32): lanes 0-15 -> rows (K0,K1);
            //                       lanes 16-31 -> rows (K2,K3)
            v2f Bd1, Bd2, Bd3, Bd4, Bd5, Bd6, Bu, Bv;
            Bd1.x = hi ? abz : abx;  Bd1.y = hi ? nc1 : aby;
            Bd2.x = hi ? acz : acx;  Bd2.y = hi ? nc2 : acy;
            Bd3.x = hi ? abz : abx;  Bd3.y = hi ? nc3 : aby;
            Bd4.x = hi ? acz : acx;  Bd4.y = hi ? nc4 : acy;
            Bd5.x = hi ? abz : abx;  Bd5.y = hi ? nc5 : aby;
            Bd6.x = hi ? acz : acx;  Bd6.y = hi ? nc6 : acy;
            Bu.x  = hi ? 0.f : re1;  Bu.y  = hi ? rcu : re2;
            Bv.x  = hi ? 0.f : rf1;  Bv.y  = hi ? rcv : rf2;

            v8f Z = {};
            // D[m] on lane l = value for (point m+8*(l>=16), triangle l%16)
            v8f D1 = __builtin_amdgcn_wmma_f32_16x16x4_f32(false, A, false, Bd1, (short)0, Z, false, false);
            v8f D2 = __builtin_amdgcn_wmma_f32_16x16x4_f32(false, A, false, Bd2, (short)0, Z, false, false);
            v8f D3 = __builtin_amdgcn_wmma_f32_16x16x4_f32(false, A, false, Bd3, (short)0, Z, false, false);
            v8f D4 = __builtin_amdgcn_wmma_f32_16x16x4_f32(false, A, false, Bd4, (short)0, Z, false, false);
            v8f D5 = __builtin_amdgcn_wmma_f32_16x16x4_f32(false, A, false, Bd5, (short)0, Z, false, false);
            v8f D6 = __builtin_amdgcn_wmma_f32_16x16x4_f32(false, A, false, Bd6, (short)0, Z, false, false);
            v8f UN = __builtin_amdgcn_wmma_f32_16x16x4_f32(false, A, false, Bu,  (short)0, Z, false, false);
            v8f VN = __builtin_amdgcn_wmma_f32_16x16x4_f32(false, A, false, Bv,  (short)0, Z, false, false);

#pragma unroll
            for (int m = 0; m < 8; ++m) {
                const float px = pxs[m], py = pys[m], pz = pzs[m];
                const float d1 = D1[m], d2 = D2[m], d3 = D3[m];
                const float d4 = D4[m], d5 = D5[m], d6 = D6[m];

                const float va = d3 * d6 - d5 * d4;
                const float vb = d5 * d2 - d1 * d6;
                const float vc = d1 * d4 - d3 * d2;
                float den = va + vb + vc;
                den = (fabsf(den) < EPSF) ? EPSF : den;
                const float rden = frcp(den);
                const float vi = vb * rden, wi = vc * rden;
                float ptx = ax + vi * abx + wi * acx;
                float pty = ay + vi * aby + wi * acy;
                float ptz = az + vi * abz + wi * acz;
                // edge BC
                const float d43 = d4 - d3, d56 = d5 - d6;
                const float tbc = d43 * frcp(fmaxf(d43 + d56, EPSF));
                bool c0 = (va <= 0.f) && (d43 >= 0.f) && (d56 >= 0.f);
                ptx = c0 ? (bx + tbc * bcx) : ptx;
                pty = c0 ? (by + tbc * bcy) : pty;
                ptz = c0 ? (bz + tbc * bcz) : ptz;
                // edge AC
                const float tac = d2 * frcp(fmaxf(d2 - d6, EPSF));
                c0 = (vb <= 0.f) && (d2 >= 0.f) && (d6 <= 0.f);
                ptx = c0 ? (ax + tac * acx) : ptx;
                pty = c0 ? (ay + tac * acy) : pty;
                ptz = c0 ? (az + tac * acz) : ptz;
                // edge AB
                const float tab = d1 * frcp(fmaxf(d1 - d3, EPSF));
                c0 = (vc <= 0.f) && (d1 >= 0.f) && (d3 <= 0.f);
                ptx = c0 ? (ax + tab * abx) : ptx;
                pty = c0 ? (ay + tab * aby) : pty;
                ptz = c0 ? (az + tab * abz) : ptz;
                // vertices C, B, A (in reference override order)
                c0 = (d6 >= 0.f) && (d5 <= d6);
                ptx = c0 ? cx : ptx; pty = c0 ? cy : pty; ptz = c0 ? cz : ptz;
                c0 = (d3 >= 0.f) && (d4 <= d3);
                ptx = c0 ? bx : ptx; pty = c0 ? by : pty; ptz = c0 ? bz : ptz;
                c0 = (d1 <= 0.f) && (d2 <= 0.f);
                ptx = c0 ? ax : ptx; pty = c0 ? ay : pty; ptz = c0 ? az : ptz;

                const float dx = px - ptx, dy = py - pty, dz = pz - ptz;
                md2[m] = fminf(md2[m], dx * dx + dy * dy + dz * dz);

                // +z ray crossing (numerators from WMMA)
                const float u = UN[m] * riv;
                const float v = VN[m] * riv;
                const float w = 1.0f - u - v;
                const bool in2d = (u >= 0.f) && (v >= 0.f) && (w >= 0.f) && (rnd > 0.5f);
                const float zhit = u * az + v * bz + w * cz;
                par[m] += (in2d && (zhit > pz)) ? 1 : 0;
            }
        }
    }

    // -------- reduce across 16 triangle columns (stays in lane halves) ----
#pragma unroll
    for (int m = 0; m < 8; ++m) {
        float dmin = md2[m];
        int   psum = par[m];
#pragma unroll
        for (int mask = 1; mask <= 8; mask <<= 1) {
            dmin  = fminf(dmin, __shfl_xor(dmin, mask, 32));
            psum += __shfl_xor(psum, mask, 32);
        }
        if (n16 == 0) {
            int pid = tileBase + m + (hi ? 8 : 0);
            if (pid < P)
                out[pid] = (psum & 1) ? sqrtf(dmin) : 0.0f;
        }
    }
}

extern "C" void kernel_launch(void* const* d_in, const int* in_sizes, int n_in,
                              void* d_out, int out_size, void* d_ws, size_t ws_size,
                              hipStream_t stream) {
    (void)d_ws; (void)ws_size; (void)n_in;
    const int*   faces = (const int*)d_in[0];
    const float* verts = (const float*)d_in[1];
    float*       out   = (float*)d_out;

    const int F = in_sizes[0] / 3;      // (1,F,3) int32
    const int V = in_sizes[1] / 3;      // (1,V,3) f32

    // G from out_size (B==1): integer cube root, no device reads (graph-safe).
    int G = 1;
    while ((long long)(G + 1) * (G + 1) * (G + 1) <= (long long)out_size) ++G;
    const int P = G * G * G;

    const int blocks = (P + 127) / 128;   // 128 points (8 wave-tiles) per block
    sdf_wmma_kernel<<<blocks, TPB, 0, stream>>>(faces, verts, out, F, V, G, P);
}